// SSM_21689584845041
// MI455X (gfx1250) — compile-verified
//
#include <hip/hip_runtime.h>
#include <math.h>

// ---------------------------------------------------------------------------
// SSM scan for MI455X (gfx1250, wave32).
//
// Sizes (fixed by the reference):
//   D_MODEL=128, D_STATE=16, BATCH=4, SEQ=4096.
//
// Roofline: total HBM traffic ~16 MB (x in, y out) + ~11 MB L2-resident
// workspace -> ~1 us at 23.3 TB/s.  The cost is ~200-300M v_exp/v_log in the
// scan, so the kernel is transcendental-VALU bound.  We therefore:
//   * run the three x-projections as one WMMA f32 16x16x4 GEMM pass,
//   * chunk the 4096-long scan into 16 chunks (exact two-level prefix),
//     giving 4096 waves of parallelism instead of 256.
// ---------------------------------------------------------------------------

#define D_MODEL 128
#define D_STATE 16
#define BATCH   4
#define SEQ     4096
#define NCHUNK  16
#define CHUNK   (SEQ / NCHUNK)   // 256
#define LOG_CLAMP 70.0f

typedef float v2f __attribute__((ext_vector_type(2)));
typedef float v8f __attribute__((ext_vector_type(8)));

__device__ __forceinline__ float softplus_f(float z) {
    // numerically stable softplus: max(z,0) + log1p(exp(-|z|))
    return fmaxf(z, 0.0f) + log1pf(__expf(-fabsf(z)));
}

// ---------------------------------------------------------------------------
// Kernel 1: projections via V_WMMA_F32_16X16X4_F32.
//   Bm  = x @ W_B + b_B   [B*S, 16]
//   Cc  = x @ W_C + b_C   [B*S, 16]
//   xwd = x @ W_D + b_D   [B*S]     (W_D padded to a 16-col tile, col 0 used)
// One wave handles a 16-row tile of M = B*S; K-loop of 32 steps of K=4.
// A 16x4 fragment: lanes 0-15 hold row M=lane, K={0,1}; lanes 16-31 K={2,3}.
// B 4x16 fragment: lanes 0-15 hold N=lane for K={0,1}; lanes 16-31 K={2,3}.
// C/D 16x16: VGPR r -> M = r + 8*(lane>=16), N = lane&15.
// The W_D column-pad is branchless (broadcast load * lane mask) so the K-loop
// contains no exec-mask manipulation: WMMA requires EXEC all-1s anyway.
// ---------------------------------------------------------------------------
__global__ void proj_wmma_kernel(const float* __restrict__ x,
                                 const float* __restrict__ W_B, const float* __restrict__ b_B,
                                 const float* __restrict__ W_C, const float* __restrict__ b_C,
                                 const float* __restrict__ W_D, const float* __restrict__ b_D,
                                 float* __restrict__ Bm, float* __restrict__ Cc,
                                 float* __restrict__ xwd) {
    const int lane  = threadIdx.x & 31;
    const int wave  = threadIdx.x >> 5;
    const int tile  = blockIdx.x * (blockDim.x >> 5) + wave;  // 16-row tile index
    const int half  = lane >> 4;          // 0: lanes 0-15, 1: lanes 16-31
    const int lid   = lane & 15;
    const int row   = tile * 16 + lid;    // A-matrix row owned by this lane
    const int khalf = half * 2;           // K sub-offset {0,2}
    const float dmask = (lid == 0) ? 1.0f : 0.0f;  // keep only column 0 of the W_D tile

    v8f accB = {}; v8f accC = {}; v8f accD = {};
    const float* xrow = x + (size_t)row * D_MODEL;

    for (int k0 = 0; k0 < D_MODEL; k0 += 4) {
        const v2f a = *(const v2f*)(xrow + k0 + khalf);  // 8B-aligned -> global_load_b64
        v2f fB, fC, fD;
        fB.x = W_B[(k0 + khalf + 0) * D_STATE + lid];
        fB.y = W_B[(k0 + khalf + 1) * D_STATE + lid];
        fC.x = W_C[(k0 + khalf + 0) * D_STATE + lid];
        fC.y = W_C[(k0 + khalf + 1) * D_STATE + lid];
        // branchless pad: uniform broadcast load, zeroed in lanes n!=0
        fD.x = W_D[k0 + khalf + 0] * dmask;
        fD.y = W_D[k0 + khalf + 1] * dmask;
        accB = __builtin_amdgcn_wmma_f32_16x16x4_f32(false, a, false, fB, (short)0, accB, false, false);
        accC = __builtin_amdgcn_wmma_f32_16x16x4_f32(false, a, false, fC, (short)0, accC, false, false);
        accD = __builtin_amdgcn_wmma_f32_16x16x4_f32(false, a, false, fD, (short)0, accD, false, false);
    }

    const float bBn = b_B[lid];
    const float bCn = b_C[lid];
    const float bDs = b_D[0];
#pragma unroll
    for (int r = 0; r < 8; ++r) {
        const int orow = tile * 16 + r + 8 * half;
        Bm[(size_t)orow * D_STATE + lid] = accB[r] + bBn;
        Cc[(size_t)orow * D_STATE + lid] = accC[r] + bCn;
        if (lid == 0) xwd[orow] = accD[r] + bDs;
    }
}

// ---------------------------------------------------------------------------
// Kernel 2: disc[b,s,d] = softplus(P[d] + xwd[b,s]) and per-chunk sums of disc
// (for the two-level prefix over s).  Thread = (b, ch, d), d fastest
// (coalesced disc writes; xwd reads broadcast across d).
// ---------------------------------------------------------------------------
__global__ void disc_kernel(const float* __restrict__ xwd, const float* __restrict__ P,
                            float* __restrict__ disc, float* __restrict__ discsum) {
    const int t  = blockIdx.x * blockDim.x + threadIdx.x;  // [0, B*NCHUNK*D)
    const int d  = t % D_MODEL;
    const int g  = t / D_MODEL;
    const int ch = g % NCHUNK;
    const int b  = g / NCHUNK;

    const float Pd = P[d];
    float acc = 0.0f;
    const int s0 = ch * CHUNK;
    for (int s = s0; s < s0 + CHUNK; ++s) {
        const float dc = softplus_f(Pd + xwd[b * SEQ + s]);
        disc[((size_t)(b * SEQ + s)) * D_MODEL + d] = dc;
        acc += dc;
    }
    discsum[(b * D_MODEL + d) * NCHUNK + ch] = acc;
}

// ---------------------------------------------------------------------------
// Kernel 3/5: in-place exclusive prefix over NCHUNK per channel (tiny).
// ---------------------------------------------------------------------------
__global__ void scan_chunks_kernel(float* __restrict__ arr, int nchannels) {
    const int t = blockIdx.x * blockDim.x + threadIdx.x;
    if (t >= nchannels) return;
    float run = 0.0f;
    float* p = arr + (size_t)t * NCHUNK;
#pragma unroll
    for (int c = 0; c < NCHUNK; ++c) { const float v = p[c]; p[c] = run; run += v; }
}

// ---------------------------------------------------------------------------
// Kernel 4: per-chunk sums of AinvsB (the clamped scan increment).
// Thread = (b, d, ch, n), n in the low 4 lane bits.
// Uses cumsum(disc*A) == A*cumsum(disc) so only the scalar disc prefix is
// needed; the clamp max(L, log|Bx|-70) is applied per element as in the ref.
// ---------------------------------------------------------------------------
__global__ void scan_pass1_kernel(const float* __restrict__ x,
                                  const float* __restrict__ Araw,
                                  const float* __restrict__ disc,
                                  const float* __restrict__ discpre,
                                  const float* __restrict__ Bm,
                                  float* __restrict__ sumS) {
    const int t  = blockIdx.x * blockDim.x + threadIdx.x;  // [0, B*D*NCHUNK*16)
    const int n  = t & 15;
    const int g  = t >> 4;
    const int ch = g % NCHUNK;
    const int g2 = g / NCHUNK;
    const int d  = g2 % D_MODEL;
    const int b  = g2 / D_MODEL;

    const float A    = -softplus_f(Araw[d * D_STATE + n]);
    const float invA = 1.0f / A;
    float Dcum = discpre[(b * D_MODEL + d) * NCHUNK + ch];
    float acc  = 0.0f;
    const int s0 = ch * CHUNK;
    for (int s = s0; s < s0 + CHUNK; ++s) {
        const size_t bs = (size_t)(b * SEQ + s);
        const float dc  = disc[bs * D_MODEL + d];
        Dcum += dc;
        const float L   = A * Dcum;
        const float xv  = x[bs * D_MODEL + d];
        const float bm  = Bm[bs * D_STATE + n];
        const float Bx  = bm * invA * (__expf(dc * A) - 1.0f) * xv;
        const float bxl = __logf(fabsf(Bx));
        const float Lc  = fmaxf(L, bxl - LOG_CLAMP);
        const float sgn = (Bx > 0.0f) ? 1.0f : ((Bx < 0.0f) ? -1.0f : 0.0f);
        acc += __expf(bxl - Lc) * sgn;
    }
    sumS[((b * D_MODEL + d) * D_STATE + n) * NCHUNK + ch] = acc;
}

// ---------------------------------------------------------------------------
// Kernel 6: final pass.  Recompute increments with both prefix offsets,
// h = Ssum*exp(Lc), contract over n (16 lanes) with shfl_xor, write y.
// ---------------------------------------------------------------------------
__global__ void scan_final_kernel(const float* __restrict__ x,
                                  const float* __restrict__ Araw,
                                  const float* __restrict__ disc,
                                  const float* __restrict__ discpre,
                                  const float* __restrict__ Bm,
                                  const float* __restrict__ Cc,
                                  const float* __restrict__ sumSpre,
                                  float* __restrict__ y) {
    const int t  = blockIdx.x * blockDim.x + threadIdx.x;
    const int n  = t & 15;
    const int g  = t >> 4;
    const int ch = g % NCHUNK;
    const int g2 = g / NCHUNK;
    const int d  = g2 % D_MODEL;
    const int b  = g2 / D_MODEL;

    const float A    = -softplus_f(Araw[d * D_STATE + n]);
    const float invA = 1.0f / A;
    float Dcum = discpre[(b * D_MODEL + d) * NCHUNK + ch];
    float Ssum = sumSpre[((b * D_MODEL + d) * D_STATE + n) * NCHUNK + ch];
    const int s0 = ch * CHUNK;
    for (int s = s0; s < s0 + CHUNK; ++s) {
        const size_t bs = (size_t)(b * SEQ + s);
        const float dc  = disc[bs * D_MODEL + d];
        Dcum += dc;
        const float L   = A * Dcum;
        const float xv  = x[bs * D_MODEL + d];
        const float bm  = Bm[bs * D_STATE + n];
        const float Bx  = bm * invA * (__expf(dc * A) - 1.0f) * xv;
        const float bxl = __logf(fabsf(Bx));
        const float Lc  = fmaxf(L, bxl - LOG_CLAMP);
        const float sgn = (Bx > 0.0f) ? 1.0f : ((Bx < 0.0f) ? -1.0f : 0.0f);
        Ssum += __expf(bxl - Lc) * sgn;
        const float h = Ssum * __expf(Lc);
        float contrib = h * Cc[bs * D_STATE + n];
        // reduce over the 16 state lanes (wave32: xor masks < 16 stay in-group)
        contrib += __shfl_xor(contrib, 1, 32);
        contrib += __shfl_xor(contrib, 2, 32);
        contrib += __shfl_xor(contrib, 4, 32);
        contrib += __shfl_xor(contrib, 8, 32);
        if (n == 0) y[bs * D_MODEL + d] = contrib;
    }
}

// ---------------------------------------------------------------------------
extern "C" void kernel_launch(void* const* d_in, const int* in_sizes, int n_in,
                              void* d_out, int out_size, void* d_ws, size_t ws_size,
                              hipStream_t stream) {
    (void)in_sizes; (void)n_in; (void)out_size; (void)ws_size;
    const float* x    = (const float*)d_in[0];
    const float* Araw = (const float*)d_in[1];
    const float* P    = (const float*)d_in[2];
    const float* W_B  = (const float*)d_in[3];
    const float* b_B  = (const float*)d_in[4];
    const float* W_C  = (const float*)d_in[5];
    const float* b_C  = (const float*)d_in[6];
    const float* W_D  = (const float*)d_in[7];
    const float* b_D  = (const float*)d_in[8];
    float* y = (float*)d_out;

    // workspace layout (bytes); every buffer is fully rewritten each call
    constexpr size_t BS       = (size_t)BATCH * SEQ;          // 16384
    constexpr size_t szBm     = BS * D_STATE * 4;             // 1 MB
    constexpr size_t szCc     = BS * D_STATE * 4;             // 1 MB
    constexpr size_t szXwd    = BS * 4;                       // 64 KB
    constexpr size_t szDisc   = BS * D_MODEL * 4;             // 8 MB
    constexpr size_t szDsum   = (size_t)BATCH * D_MODEL * NCHUNK * 4;            // 32 KB
    char* ws = (char*)d_ws;
    float* Bm   = (float*)(ws);
    float* Cc   = (float*)(ws + szBm);
    float* xwd  = (float*)(ws + szBm + szCc);
    float* disc = (float*)(ws + szBm + szCc + szXwd);
    float* dsum = (float*)(ws + szBm + szCc + szXwd + szDisc);
    float* sumS = (float*)(ws + szBm + szCc + szXwd + szDisc + szDsum);

    // 1. WMMA projections: 1024 tiles of 16 rows, 8 waves per 256-thread block
    proj_wmma_kernel<<<(BS / 16) / 8, 256, 0, stream>>>(x, W_B, b_B, W_C, b_C, W_D, b_D,
                                                        Bm, Cc, xwd);
    // 2. disc + per-chunk disc sums
    disc_kernel<<<(BATCH * D_MODEL * NCHUNK) / 256, 256, 0, stream>>>(xwd, P, disc, dsum);
    // 3. exclusive chunk-prefix of disc sums (512 channels)
    scan_chunks_kernel<<<(BATCH * D_MODEL + 255) / 256, 256, 0, stream>>>(dsum, BATCH * D_MODEL);
    // 4. per-chunk sums of the clamped scan increments (4096 waves)
    scan_pass1_kernel<<<(BATCH * D_MODEL * D_STATE * NCHUNK) / 256, 256, 0, stream>>>(
        x, Araw, disc, dsum, Bm, sumS);
    // 5. exclusive chunk-prefix of increment sums (8192 channels)
    scan_chunks_kernel<<<(BATCH * D_MODEL * D_STATE + 255) / 256, 256, 0, stream>>>(
        sumS, BATCH * D_MODEL * D_STATE);
    // 6. final pass: h, contraction over N, y
    scan_final_kernel<<<(BATCH * D_MODEL * D_STATE * NCHUNK) / 256, 256, 0, stream>>>(
        x, Araw, disc, dsum, Bm, Cc, sumS, y);
}